// Encoder_sparse_54125177864775
// MI455X (gfx1250) — compile-verified
//
#include <hip/hip_runtime.h>
#include <math.h>

#define NROWS 20000
#define KIN   1024
#define DOUT  64
#define NNZE  320000

typedef __attribute__((ext_vector_type(2))) float v2f;
typedef __attribute__((ext_vector_type(8))) float v8f;

// D = A(16x4) * B(4x16) + C, fp32 WMMA (CDNA5, wave32)
__device__ __forceinline__ v8f wmma4(v2f a, v2f b, v8f c) {
  return __builtin_amdgcn_wmma_f32_16x16x4_f32(false, a, false, b, (short)0, c,
                                               false, false);
}

// Stage a 64(K) x 64(N) tile of B into LDS with K-pair interleave:
//   Bs[p*128 + n*2 + j] = B[2p + j][n],  p in [0,32), n in [0,64), j in {0,1}
// so a WMMA B-fragment {B[ka][n], B[ka+1][n]} is one contiguous ds_load_b64.
__device__ __forceinline__ void stage_B64x64(float* __restrict__ Bs,
                                             const float* __restrict__ B,
                                             int ldb, int tid, int nthreads) {
  for (int q = tid; q < 32 * 32; q += nthreads) {
    const int p = q >> 5;
    const int n = (q & 31) * 2;
    const float2 r0 = *(const float2*)(B + (size_t)(2 * p) * ldb + n);
    const float2 r1 = *(const float2*)(B + (size_t)(2 * p + 1) * ldb + n);
    float4 w;
    w.x = r0.x; w.y = r1.x; w.z = r0.y; w.w = r1.y;
    *(float4*)(Bs + p * 128 + n * 2) = w;
  }
}

// -------- C = A[nrows,1024] @ B[1024,64], per-wave 16x64 tile --------
__global__ __launch_bounds__(128)
void gemm_k1024(const float* __restrict__ A, const float* __restrict__ B,
                float* __restrict__ Out, int nrows) {
  __shared__ float Bs[64 * DOUT];  // 16 KB interleaved chunk of W1
  const int wave = threadIdx.x >> 5, lane = threadIdx.x & 31;
  const int tile = blockIdx.x * 4 + wave;
  const bool active = (tile * 16) < nrows;
  const int half = lane >> 4, lm = lane & 15;
  const float* Arow = A + (size_t)((active ? tile : 0) * 16 + lm) * KIN;
  v8f acc0 = {}, acc1 = {}, acc2 = {}, acc3 = {};
  for (int kc = 0; kc < KIN; kc += 64) {
    stage_B64x64(Bs, B + (size_t)kc * DOUT, DOUT, threadIdx.x, blockDim.x);
    __syncthreads();
    if (active) {  // wave-uniform: EXEC all-ones inside
#pragma unroll
      for (int k0 = 0; k0 < 64; k0 += 4) {
        const v2f a = *(const v2f*)(Arow + kc + k0 + half * 2);
        const float* Bp = Bs + ((k0 >> 1) + half) * 128 + lm * 2;
        const v2f b0 = *(const v2f*)(Bp);
        const v2f b1 = *(const v2f*)(Bp + 32);
        const v2f b2 = *(const v2f*)(Bp + 64);
        const v2f b3 = *(const v2f*)(Bp + 96);
        acc0 = wmma4(a, b0, acc0);
        acc1 = wmma4(a, b1, acc1);
        acc2 = wmma4(a, b2, acc2);
        acc3 = wmma4(a, b3, acc3);
      }
    }
    __syncthreads();
  }
  if (active) {
    float* O = Out + (size_t)(tile * 16 + half * 8) * DOUT + lm;
#pragma unroll
    for (int r = 0; r < 8; ++r) {
      O[(size_t)r * DOUT + 0]  = acc0[r];
      O[(size_t)r * DOUT + 16] = acc1[r];
      O[(size_t)r * DOUT + 32] = acc2[r];
      O[(size_t)r * DOUT + 48] = acc3[r];
    }
  }
}

// -------- C = A[nrows,64] @ B[64,1024]; mode0: fused MeanAct/DispAct --------
__global__ __launch_bounds__(128)
void gemm_k64(const float* __restrict__ A, const float* __restrict__ B,
              float* __restrict__ out0, float* __restrict__ out1, int mode,
              int nrows) {
  __shared__ float Bs[64 * 64];  // 16 KB: interleaved B[:, cg*64 .. cg*64+63]
  const int cg = blockIdx.y;
  stage_B64x64(Bs, B + (size_t)cg * 64, KIN, threadIdx.x, blockDim.x);
  __syncthreads();
  const int wave = threadIdx.x >> 5, lane = threadIdx.x & 31;
  const int tile = blockIdx.x * 4 + wave;
  if ((tile * 16) >= nrows) return;  // after the only barrier
  const int half = lane >> 4, lm = lane & 15;
  const float* Arow = A + (size_t)(tile * 16 + lm) * DOUT;
  v8f acc0 = {}, acc1 = {}, acc2 = {}, acc3 = {};
#pragma unroll
  for (int k0 = 0; k0 < 64; k0 += 4) {
    const v2f a = *(const v2f*)(Arow + k0 + half * 2);
    const float* Bp = Bs + ((k0 >> 1) + half) * 128 + lm * 2;
    const v2f b0 = *(const v2f*)(Bp);
    const v2f b1 = *(const v2f*)(Bp + 32);
    const v2f b2 = *(const v2f*)(Bp + 64);
    const v2f b3 = *(const v2f*)(Bp + 96);
    acc0 = wmma4(a, b0, acc0);
    acc1 = wmma4(a, b1, acc1);
    acc2 = wmma4(a, b2, acc2);
    acc3 = wmma4(a, b3, acc3);
  }
#pragma unroll
  for (int r = 0; r < 8; ++r) {
    const size_t base =
        (size_t)(tile * 16 + half * 8 + r) * KIN + (size_t)cg * 64 + lm;
    float v[4] = {acc0[r], acc1[r], acc2[r], acc3[r]};
#pragma unroll
    for (int nb = 0; nb < 4; ++nb) {
      const size_t o = base + nb * 16;
      if (mode == 0) {
        const float x = v[nb];
        out0[o] = fminf(fmaxf(expf(x), 1e-5f), 1e6f);       // MeanAct
        const float sp = (x > 20.f) ? x : log1pf(expf(x));  // softplus
        out1[o] = fminf(fmaxf(sp, 1e-4f), 1e4f);            // DispAct
      } else {
        out0[o] = v[nb];
      }
    }
  }
}

// -------- COO SpMM scatter: Y[row[e], :] += vals[e] * X[col[e], :] --------
__global__ void spmm_scatter(const int* __restrict__ rows,
                             const int* __restrict__ cols,
                             const float* __restrict__ vals,
                             const float* __restrict__ X, float* __restrict__ Y,
                             long long total, int shift, int d) {
  const long long gid = (long long)blockIdx.x * blockDim.x + threadIdx.x;
  if (gid >= total) return;
  const long long e = gid >> shift;
  const int c = (int)(gid & ((1LL << shift) - 1)) * 4;
  const int r = rows[e], cl = cols[e];
  const float v = vals[e];
  const float4 x = *(const float4*)(X + (size_t)cl * d + c);
  float* y = Y + (size_t)r * d + c;
  atomicAdd(y + 0, v * x.x);
  atomicAdd(y + 1, v * x.y);
  atomicAdd(y + 2, v * x.z);
  atomicAdd(y + 3, v * x.w);
}

// -------- ret[i] = relu(Hpl[i,:]) @ W @ relu(Cx[i,:]) + bias --------
__global__ __launch_bounds__(128)
void bilinear_dot(const float* __restrict__ Hpl, const float* __restrict__ Cx,
                  const float* __restrict__ W, const float* __restrict__ bias,
                  float* __restrict__ out, int n) {
  const int r = blockIdx.x * 4 + (threadIdx.x >> 5);
  const int lane = threadIdx.x & 31;
  if (r >= n) return;
  const float h0 = fmaxf(Hpl[(size_t)r * 64 + lane], 0.f);
  const float h1 = fmaxf(Hpl[(size_t)r * 64 + lane + 32], 0.f);
  float acc = 0.f;
  for (int k = 0; k < 64; ++k) {
    const float c = fmaxf(Cx[(size_t)r * 64 + k], 0.f);
    acc = fmaf(h0 * W[lane * 64 + k] + h1 * W[(lane + 32) * 64 + k], c, acc);
  }
#pragma unroll
  for (int off = 16; off > 0; off >>= 1) acc += __shfl_down(acc, off, 32);
  if (lane == 0) out[r] = acc + bias[0];
}

__global__ void fill_f32(float* __restrict__ p, long long n) {
  const long long i = (long long)blockIdx.x * blockDim.x + threadIdx.x;
  if (i < n) p[i] = 0.f;
}

extern "C" void kernel_launch(void* const* d_in, const int* in_sizes, int n_in,
                              void* d_out, int out_size, void* d_ws,
                              size_t ws_size, hipStream_t stream) {
  const float* feat   = (const float*)d_in[0];
  const float* feat_a = (const float*)d_in[1];
  const int*   row    = (const int*)d_in[2];
  const int*   col    = (const int*)d_in[3];
  const float* vals   = (const float*)d_in[4];
  const int*   row_a  = (const int*)d_in[5];
  const int*   col_a  = (const int*)d_in[6];
  const float* vals_a = (const float*)d_in[7];
  const float* W1     = (const float*)d_in[8];
  const float* W2     = (const float*)d_in[9];
  const float* discW  = (const float*)d_in[10];
  const float* discB  = (const float*)d_in[11];

  // Output layout: hiden_emb[N,64] | h[N,1024] | ret[N] | ret_a[N] | mean[N,1024] | disp[N,1024]
  float* z    = (float*)d_out;
  float* h    = z + (size_t)NROWS * DOUT;
  float* ret  = h + (size_t)NROWS * KIN;
  float* reta = ret + NROWS;
  float* mean = reta + NROWS;
  float* disp = mean + (size_t)NROWS * KIN;

  // Workspace: z0 | za0 | embA | embS | zW2  (≈102 MB)
  float* z0   = (float*)d_ws;
  float* za0  = z0 + (size_t)NROWS * DOUT;
  float* embA = za0 + (size_t)NROWS * DOUT;
  float* embS = embA + (size_t)NROWS * DOUT;
  float* zW2  = embS + (size_t)NROWS * DOUT;

  const int rtiles = (NROWS + 15) / 16;  // 1250
  const int gblk   = (rtiles + 3) / 4;   // 313 blocks of 4 waves

  // 1. zero the atomic-accumulated buffers
  {
    const long long nz = (long long)NROWS * DOUT;
    const long long nh = (long long)NROWS * KIN;
    fill_f32<<<(unsigned)((nz + 255) / 256), 256, 0, stream>>>(z, nz);
    fill_f32<<<(unsigned)((nh + 255) / 256), 256, 0, stream>>>(h, nh);
    fill_f32<<<(unsigned)((nz + 255) / 256), 256, 0, stream>>>(embA, nz);
    fill_f32<<<(unsigned)((nz + 255) / 256), 256, 0, stream>>>(embS, nz);
  }

  // 2. z0 = feat @ W1 ; za0 = feat_a @ W1   (fp32 WMMA)
  gemm_k1024<<<gblk, 128, 0, stream>>>(feat, W1, z0, NROWS);
  gemm_k1024<<<gblk, 128, 0, stream>>>(feat_a, W1, za0, NROWS);

  // 3. mean/disp = acts(z0 @ W2)   (fp32 WMMA, fused ZINB heads)
  gemm_k64<<<dim3(gblk, KIN / 64), 128, 0, stream>>>(z0, W2, mean, disp, 0,
                                                     NROWS);

  // 4. SpMMs over d=64
  {
    const long long tot = (long long)NNZE * (DOUT / 4);
    const unsigned nb = (unsigned)((tot + 255) / 256);
    spmm_scatter<<<nb, 256, 0, stream>>>(row, col, vals, z0, z, tot, 4, DOUT);
    spmm_scatter<<<nb, 256, 0, stream>>>(row, col, vals, za0, embA, tot, 4,
                                         DOUT);
    spmm_scatter<<<nb, 256, 0, stream>>>(row_a, col_a, vals_a, za0, embS, tot,
                                         4, DOUT);
  }

  // 5. zW2 = z @ W2, then h = spmm(A, zW2)
  gemm_k64<<<dim3(gblk, KIN / 64), 128, 0, stream>>>(z, W2, zW2, zW2, 1, NROWS);
  {
    const long long tot = (long long)NNZE * (KIN / 4);
    const unsigned nb = (unsigned)((tot + 255) / 256);
    spmm_scatter<<<nb, 256, 0, stream>>>(row, col, vals, zW2, h, tot, 8, KIN);
  }

  // 6. discriminator heads (ReLU applied at read time)
  {
    const unsigned nb = (NROWS + 3) / 4;
    bilinear_dot<<<nb, 128, 0, stream>>>(embA, z, discW, discB, ret, NROWS);
    bilinear_dot<<<nb, 128, 0, stream>>>(embS, embA, discW, discB, reta, NROWS);
  }
}